// LLamaAttention_30167850287807
// MI455X (gfx1250) — compile-verified
//
#include <hip/hip_runtime.h>
#include <hip/hip_bf16.h>

// ---------------- problem constants (match reference) ----------------
#define B_      32
#define H_      32
#define HKV_    8
#define G_      4           // H / HKV
#define D_      128
#define N_      4096
#define BLOCK_  16
#define NB_     256         // N / BLOCK
#define SPLIT_  8           // context splits per (b, kv-head)
#define CHUNK_  (N_ / SPLIT_)   // 512 positions per wave
#define TILES_  (CHUNK_ / 16)   // 32 tiles of 16 positions
#define SCALE_  0.08838834764831845f
#define PSLOT_  (G_ * D_ + 2 * G_)   // 520 floats per partial slot

typedef float v2f __attribute__((ext_vector_type(2)));
typedef float v8f __attribute__((ext_vector_type(8)));

// ---------------- kernel 0: RoPE for Q and current K ----------------
__global__ void rope_kernel(const float* __restrict__ Q, const float* __restrict__ K,
                            const float* __restrict__ cosb, const float* __restrict__ sinb,
                            float* __restrict__ Qr, float* __restrict__ Kr) {
  const int idx = blockIdx.x * blockDim.x + threadIdx.x;
  const int nq = B_ * H_ * D_;
  const int nk = B_ * HKV_ * D_;
  if (idx < nq) {
    const int b = idx / (H_ * D_);
    const int d = idx % D_;
    const float x  = Q[idx];
    const float xr = (d < D_ / 2) ? -Q[idx + D_ / 2] : Q[idx - D_ / 2];
    Qr[idx] = x * cosb[b * D_ + d] + xr * sinb[b * D_ + d];
  } else if (idx < nq + nk) {
    const int i = idx - nq;
    const int b = i / (HKV_ * D_);
    const int d = i % D_;
    const float x  = K[i];
    const float xr = (d < D_ / 2) ? -K[i + D_ / 2] : K[i - D_ / 2];
    Kr[i] = x * cosb[b * D_ + d] + xr * sinb[b * D_ + d];
  }
}

// ---------------- kernel 1: flash-decode partials (1 wave per split) ----------------
__global__ __launch_bounds__(128) void flash_partial_kernel(
    const float* __restrict__ Qr, const float* __restrict__ Kr,
    const float* __restrict__ Vcur,
    const float* __restrict__ Kcache, const float* __restrict__ Vcache,
    const int* __restrict__ block_tables,
    float* __restrict__ part) {
  __shared__ float qlds[4][16 * D_];          // per-wave padded Q tile (16x128)

  const int lane  = threadIdx.x & 31;
  const int wib   = threadIdx.x >> 5;
  const int wid   = blockIdx.x * 4 + wib;
  const int s     = wid % SPLIT_;
  const int kh    = (wid / SPLIT_) % HKV_;
  const int b     = wid / (SPLIT_ * HKV_);
  const int row16 = lane & 15;
  const int hi    = lane >> 4;

  // stage Q^T source: heads 0..3 -> q-heads g*HKV+kh; rows 4..15 zero padding
  float* ql = qlds[wib];
  for (int i = lane; i < 16 * D_; i += 32) {
    const int r = i >> 7, d = i & (D_ - 1);
    float v = 0.0f;
    if (r < G_) v = Qr[((size_t)b * H_ + (size_t)(r * HKV_ + kh)) * D_ + d];
    ql[i] = v;
  }
  __syncthreads();

  const float* krcur = Kr   + ((size_t)b * HKV_ + kh) * D_;
  const float* vccur = Vcur + ((size_t)b * HKV_ + kh) * D_;

  v8f acc[8] = {};            // acc^T: [dim-chunk][16 dims x 16 heads]
  float m = -1e30f, l = 0.0f; // per-head (lane column) softmax state

  for (int tt = 0; tt < TILES_; ++tt) {
    const int p0 = s * CHUNK_ + tt * 16;
    const bool lastTile = (p0 + 16 == N_);       // contains current token at pos N-1
    const int bt = block_tables[b * NB_ + (p0 >> 4)];
    const size_t rowbase = (size_t)bt * BLOCK_;

    // ---- S^T = K * Q^T (16 pos x 16 heads), K frags load straight into A layout ----
    const float* krow = (lastTile && row16 == 15)
        ? krcur
        : (Kcache + (rowbase + (size_t)row16) * (HKV_ * D_) + kh * D_);

    v2f kf[32];
    #pragma unroll
    for (int kk = 0; kk < 32; ++kk)
      kf[kk] = *(const v2f*)(krow + kk * 4 + 2 * hi);

    v8f st = {};
    #pragma unroll
    for (int kk = 0; kk < 32; ++kk) {
      const v2f qf = *(const v2f*)(ql + row16 * D_ + kk * 4 + 2 * hi);  // B = Q^T frag
      st = __builtin_amdgcn_wmma_f32_16x16x4_f32(false, kf[kk], false, qf,
                                                 (short)0, st, false, false);
    }

    // ---- online softmax; head state is per-lane (replicated in lane^16) ----
    float p[8];
    float mt = -1e30f;
    #pragma unroll
    for (int r = 0; r < 8; ++r) { st[r] *= SCALE_; mt = fmaxf(mt, st[r]); }
    mt = fmaxf(mt, __shfl_xor(mt, 16, 32));
    const float mnew  = fmaxf(m, mt);
    const float alpha = __expf(m - mnew);
    float lsum = 0.0f;
    #pragma unroll
    for (int r = 0; r < 8; ++r) { p[r] = __expf(st[r] - mnew); lsum += p[r]; }
    lsum += __shfl_xor(lsum, 16, 32);
    l = l * alpha + lsum;
    m = mnew;
    #pragma unroll
    for (int ch = 0; ch < 8; ++ch) {
      #pragma unroll
      for (int r = 0; r < 8; ++r) acc[ch][r] *= alpha;
    }

    // ---- build P^T B-fragments (pos chunks of 4) via half-wave shuffles ----
    v2f bf[4];
    #pragma unroll
    for (int j = 0; j < 4; ++j) {
      const int rA = (j & 1) * 4;   // register holding pos 4j   in half hi=0
      const int rB = rA + 2;        // register holding pos 4j+2 in half hi=1
      if (j < 2) {                  // positions 0..7 live in half 0 registers
        const float tx = __shfl_xor(p[rB], 16, 32);
        const float ty = __shfl_xor(p[rB + 1], 16, 32);
        bf[j].x = hi ? tx : p[rA];
        bf[j].y = hi ? ty : p[rA + 1];
      } else {                      // positions 8..15 live in half 1 registers
        const float tx = __shfl_xor(p[rA], 16, 32);
        const float ty = __shfl_xor(p[rA + 1], 16, 32);
        bf[j].x = hi ? p[rB]     : tx;
        bf[j].y = hi ? p[rB + 1] : ty;
      }
    }

    // ---- acc^T += V^T * P^T ----
    const float* vrow[8];
    #pragma unroll
    for (int j = 0; j < 4; ++j) {
      #pragma unroll
      for (int sl = 0; sl < 2; ++sl) {
        const int pos = 4 * j + 2 * hi + sl;
        vrow[2 * j + sl] = (lastTile && pos == 15)
            ? vccur
            : (Vcache + (rowbase + (size_t)pos) * (HKV_ * D_) + kh * D_);
      }
    }
    #pragma unroll
    for (int ch = 0; ch < 8; ++ch) {
      #pragma unroll
      for (int j = 0; j < 4; ++j) {
        v2f a;                                   // A = V^T frag [16 dims x 4 pos]
        a.x = vrow[2 * j + 0][ch * 16 + row16];
        a.y = vrow[2 * j + 1][ch * 16 + row16];
        acc[ch] = __builtin_amdgcn_wmma_f32_16x16x4_f32(false, a, false, bf[j],
                                                        (short)0, acc[ch], false, false);
      }
    }
  }

  // ---- write unnormalized partials + (m, l) ----
  float* pp = part + (((size_t)b * HKV_ + kh) * SPLIT_ + s) * PSLOT_;
  const int g = row16;
  if (g < G_) {
    #pragma unroll
    for (int ch = 0; ch < 8; ++ch) {
      #pragma unroll
      for (int r = 0; r < 8; ++r)
        pp[g * D_ + ch * 16 + r + 8 * hi] = acc[ch][r];
    }
    if (!hi) { pp[G_ * D_ + g] = m; pp[G_ * D_ + G_ + g] = l; }
  }
}

// ---------------- kernel 2: log-sum-exp combine of the 8 splits ----------------
__global__ __launch_bounds__(D_) void combine_kernel(const float* __restrict__ part,
                                                     float* __restrict__ out) {
  const int x  = blockIdx.x;
  const int b  = x / (HKV_ * G_);
  const int kh = (x / G_) % HKV_;
  const int g  = x % G_;
  const int d  = threadIdx.x;
  const float* base = part + ((size_t)b * HKV_ + kh) * SPLIT_ * PSLOT_;
  float M = -1e30f;
  #pragma unroll
  for (int s = 0; s < SPLIT_; ++s)
    M = fmaxf(M, base[s * PSLOT_ + G_ * D_ + g]);
  float num = 0.0f, den = 0.0f;
  #pragma unroll
  for (int s = 0; s < SPLIT_; ++s) {
    const float w = __expf(base[s * PSLOT_ + G_ * D_ + g] - M);
    den += base[s * PSLOT_ + G_ * D_ + G_ + g] * w;
    num += base[s * PSLOT_ + g * D_ + d] * w;
  }
  out[((size_t)b * H_ + (size_t)(g * HKV_ + kh)) * D_ + d] = num / den;
}

// ---------------- host launcher ----------------
extern "C" void kernel_launch(void* const* d_in, const int* in_sizes, int n_in,
                              void* d_out, int out_size, void* d_ws, size_t ws_size,
                              hipStream_t stream) {
  const float* Q            = (const float*)d_in[0];
  const float* K            = (const float*)d_in[1];
  const float* V            = (const float*)d_in[2];
  const float* Kcache       = (const float*)d_in[3];
  const float* Vcache       = (const float*)d_in[4];
  const float* cosb         = (const float*)d_in[5];
  const float* sinb         = (const float*)d_in[6];
  const int*   block_tables = (const int*)d_in[7];
  // d_in[8] = save_slots (unused by reference output), d_in[9] = input_length (fixed 4096)

  float* ws   = (float*)d_ws;
  float* Qr   = ws;                                   // B*H*D
  float* Kr   = Qr + (size_t)B_ * H_ * D_;            // B*HKV*D
  float* part = Kr + (size_t)B_ * HKV_ * D_;          // B*HKV*SPLIT*PSLOT (~4.3 MB)
  float* out  = (float*)d_out;

  const int ropeN = B_ * H_ * D_ + B_ * HKV_ * D_;
  hipLaunchKernelGGL(rope_kernel, dim3((ropeN + 255) / 256), dim3(256), 0, stream,
                     Q, K, cosb, sinb, Qr, Kr);
  hipLaunchKernelGGL(flash_partial_kernel, dim3(B_ * HKV_ * SPLIT_ / 4), dim3(128), 0, stream,
                     Qr, Kr, V, Kcache, Vcache, block_tables, part);
  hipLaunchKernelGGL(combine_kernel, dim3(B_ * HKV_ * G_), dim3(D_), 0, stream,
                     part, out);
}